// CopyDefineNetwork_10934986736223
// MI455X (gfx1250) — compile-verified
//
#include <hip/hip_runtime.h>

typedef __attribute__((ext_vector_type(2))) float v2f;
typedef __attribute__((ext_vector_type(4))) float v4f;
typedef __attribute__((ext_vector_type(8))) float v8f;

#define BS  64
#define SEQq 512
#define Hd  768

// ---------------------------------------------------------------------------
// Phase 1: Mall[b][row][k], row = head*2+n (head: 0=cls,1=topic,2=token),
//          k = g*8+j.   Mall[b,row,k] = sum_i d0[b,g*8+i] * W_head[n, g*64+i*8+j]
// ---------------------------------------------------------------------------
__global__ void build_M_kernel(const float* __restrict__ Wsrc,
                               const float* __restrict__ Wtgt,
                               const float* __restrict__ Wc,
                               const float* __restrict__ Wt,
                               const float* __restrict__ Wk,
                               const int*   __restrict__ sid,
                               const int*   __restrict__ tid_,
                               float*       __restrict__ Mall) {
  __shared__ float d0[768];
  const int b = blockIdx.x;
  const int s = sid[b], t = tid_[b];
  for (int k = threadIdx.x; k < 384; k += blockDim.x) {
    d0[k]       = Wsrc[s * 384 + k];
    d0[384 + k] = Wtgt[t * 384 + k];
  }
  __syncthreads();

  for (int idx = threadIdx.x; idx < 6 * 768; idx += blockDim.x) {
    const int row  = idx / 768;
    const int k    = idx - row * 768;
    const int head = row >> 1, n = row & 1;
    const int g    = k >> 3,  j = k & 7;
    const float* W = (head == 0) ? Wc : ((head == 1) ? Wt : Wk);
    W += n * 6144 + g * 64 + j;
    const float* dg = &d0[g * 8];
    float acc = 0.f;
#pragma unroll
    for (int i = 0; i < 8; ++i) acc = fmaf(dg[i], W[i * 8], acc);
    Mall[(size_t)b * 4608 + idx] = acc;
  }
}

// ---------------------------------------------------------------------------
// Phase 2: per batch, gathered-GEMM  H_g[129,768] x Mall^T[768,6] via
//          v_wmma_f32_16x16x4_f32 with K-permuted b128 fragment loads.
// Row map: l==0 -> cls(h=row0); 1..64 -> topic; 65..128 -> token; 129..143 pad
// ---------------------------------------------------------------------------
__global__ __launch_bounds__(288)
void bilinear_wmma_kernel(const float* __restrict__ hidden,
                          const int*   __restrict__ topic_inds,
                          const float* __restrict__ topic_mask,
                          const int*   __restrict__ token_inds,
                          const float* __restrict__ token_mask,
                          const float* __restrict__ b_cls,
                          const float* __restrict__ b_topic,
                          const float* __restrict__ b_token,
                          const float* __restrict__ Mall,
                          float*       __restrict__ out) {
  __shared__ float Bs[16 * 768];   // zero-padded Mall[b] (N rows 6..15 = 0)
  __shared__ float Dt[9 * 256];    // one 16x16 D tile per wave
  const int b = blockIdx.x;

  for (int idx = threadIdx.x; idx < 16 * 768; idx += blockDim.x) {
    const int n = idx / 768;
    const int k = idx - n * 768;
    Bs[idx] = (n < 6) ? Mall[(size_t)b * 4608 + n * 768 + k] : 0.0f;
  }
  __syncthreads();

  const int wave  = threadIdx.x >> 5;
  const int lane  = threadIdx.x & 31;
  const int rl    = lane & 15;     // M row within tile (A) / N col (B)
  const int khalf = lane >> 4;     // 0 -> K-slots {0,1}, 1 -> K-slots {2,3}

  // resolve this lane's A row
  const int l = wave * 16 + rl;
  int pos = 0;
  if (l >= 1 && l < 65)        pos = topic_inds[b * 64 + (l - 1)];
  else if (l >= 65 && l < 129) pos = token_inds[b * 64 + (l - 65)];
  const float* rowptr = hidden + ((size_t)b * SEQq + pos) * Hd;
  __builtin_prefetch(rowptr, 0, 1);

  // K-permuted mapping: per 8 consecutive k, khalf=0 lanes own k=c..c+3,
  // khalf=1 lanes own k=c+4..c+7.  WMMA#1 takes .xy halves (quad
  // {c,c+1,c+4,c+5}), WMMA#2 takes .zw ({c+2,c+3,c+6,c+7}).  A and B use
  // the identical mapping, so the K-sum is exactly preserved.
  v8f acc = {};
  const float* aptr = rowptr + khalf * 4;
  const float* bptr = &Bs[rl * 768 + khalf * 4];
#pragma unroll 4
  for (int c = 0; c < 768; c += 8) {
    v4f av = *(const v4f*)(aptr + c);   // global_load_b128 (gathered row)
    v4f bv = *(const v4f*)(bptr + c);   // ds_load_b128
    v2f a0 = {av.x, av.y}, a1 = {av.z, av.w};
    v2f b0 = {bv.x, bv.y}, b1 = {bv.z, bv.w};
    acc = __builtin_amdgcn_wmma_f32_16x16x4_f32(false, a0, false, b0,
                                                (short)0, acc, false, false);
    acc = __builtin_amdgcn_wmma_f32_16x16x4_f32(false, a1, false, b1,
                                                (short)0, acc, false, false);
  }

  // D tile -> LDS: lane holds column N = lane&15, rows r(+8 for upper half)
  {
    float* dt = &Dt[wave * 256];
    const int N = lane & 15;
    const int mbase = (lane >> 4) * 8;
#pragma unroll
    for (int r = 0; r < 8; ++r) dt[(mbase + r) * 16 + N] = acc[r];
  }
  __syncthreads();

  // finalize: lanes 0..15 each own one row of this wave's tile
  if (lane < 16) {
    const int lrow = wave * 16 + lane;
    const float* dt = &Dt[wave * 256 + lane * 16];
    if (lrow == 0) {                                  // CLS head, softmax
      float v0 = dt[0] + b_cls[0];
      float v1 = dt[1] + b_cls[1];
      float m  = fmaxf(v0, v1);
      float e0 = __expf(v0 - m), e1 = __expf(v1 - m);
      float inv = 1.0f / (e0 + e1);
      v2f o; o.x = e0 * inv; o.y = e1 * inv;
      *(v2f*)(out + b * 2) = o;
    } else if (lrow < 65) {                           // topic head, softmax
      const int li = lrow - 1;
      float mask = topic_mask[b * 64 + li];
      float v0 = mask * dt[2] + b_topic[0];
      float v1 = mask * dt[3] + b_topic[1];
      float m  = fmaxf(v0, v1);
      float e0 = __expf(v0 - m), e1 = __expf(v1 - m);
      float inv = 1.0f / (e0 + e1);
      v2f o; o.x = e0 * inv; o.y = e1 * inv;
      *(v2f*)(out + 128 + ((size_t)b * 64 + li) * 2) = o;
    } else if (lrow < 129) {                          // token head, raw logits
      const int li = lrow - 65;
      float mask = token_mask[b * 64 + li];
      v2f o;
      o.x = mask * dt[4] + b_token[0];
      o.y = mask * dt[5] + b_token[1];
      *(v2f*)(out + 128 + 8192 + ((size_t)b * 64 + li) * 2) = o;
    }
  }
}

// ---------------------------------------------------------------------------
extern "C" void kernel_launch(void* const* d_in, const int* in_sizes, int n_in,
                              void* d_out, int out_size, void* d_ws, size_t ws_size,
                              hipStream_t stream) {
  const float* hidden     = (const float*)d_in[0];
  const float* Wsrc       = (const float*)d_in[1];
  const float* Wtgt       = (const float*)d_in[2];
  const float* W_cls      = (const float*)d_in[3];
  const float* b_cls      = (const float*)d_in[4];
  const float* W_topic    = (const float*)d_in[5];
  const float* b_topic    = (const float*)d_in[6];
  const float* W_token    = (const float*)d_in[7];
  const float* b_token    = (const float*)d_in[8];
  const int*   source_ids = (const int*)d_in[9];
  const int*   target_ids = (const int*)d_in[10];
  // d_in[11]=ent_inds, d_in[12]=ent_mask: dead in the reference outputs
  const int*   topic_inds = (const int*)d_in[13];
  const float* topic_mask = (const float*)d_in[14];
  const int*   token_inds = (const int*)d_in[15];
  const float* token_mask = (const float*)d_in[16];

  float* Mall = (float*)d_ws;       // 64 * 6 * 768 * 4 = 1.18 MB scratch
  float* out  = (float*)d_out;

  build_M_kernel<<<BS, 256, 0, stream>>>(Wsrc, Wtgt, W_cls, W_topic, W_token,
                                         source_ids, target_ids, Mall);
  bilinear_wmma_kernel<<<BS, 288, 0, stream>>>(hidden, topic_inds, topic_mask,
                                               token_inds, token_mask,
                                               b_cls, b_topic, b_token, Mall, out);
}